// TripletLoss_19301583028315
// MI455X (gfx1250) — compile-verified
//
#include <hip/hip_runtime.h>
#include <hip/hip_bf16.h>
#include <math.h>

#define BATCH 512
#define DIM   128

typedef __attribute__((ext_vector_type(2))) float v2f;
typedef __attribute__((ext_vector_type(8))) float v8f;

// ---------------------------------------------------------------------------
// Kernel 1: per-row squared norms  sq[i] = sum_d E[i,d]^2
// ---------------------------------------------------------------------------
__global__ __launch_bounds__(256) void sq_kernel(const float* __restrict__ E,
                                                 float* __restrict__ sq) {
    int i = blockIdx.x * blockDim.x + threadIdx.x;
    if (i >= BATCH) return;
    const float4* row = (const float4*)(E + i * DIM);
    float s = 0.f;
#pragma unroll
    for (int d = 0; d < DIM / 4; ++d) {
        float4 v = row[d];
        s += v.x * v.x + v.y * v.y + v.z * v.z + v.w * v.w;
    }
    sq[i] = s;
}

// ---------------------------------------------------------------------------
// Kernel 2: WMMA Gram tiles -> distance matrix
//   One wave per 16x16 tile, K swept 128/4 = 32 x v_wmma_f32_16x16x4_f32.
//   f32 A 16x4 layout: lane L -> M = L&15, VGPR pair holds K = kbase + 2*(L>>4) .. +1
//   f32 B 4x16 layout mirrors it with N = L&15 (B = E^T, so read E[col] rows).
//   C/D 16x16 f32: lane L -> N = L&15, VGPR r -> M = (L>>4)*8 + r.
// ---------------------------------------------------------------------------
__global__ __launch_bounds__(256) void gram_dist_kernel(const float* __restrict__ E,
                                                        const float* __restrict__ sq,
                                                        float* __restrict__ dist) {
    const int lane = threadIdx.x & 31;
    const int wave = threadIdx.x >> 5;
    const int tile = blockIdx.x * 8 + wave;      // 1024 tiles total (32x32)
    const int tm = (tile >> 5) << 4;             // tile row base
    const int tn = (tile & 31) << 4;             // tile col base

    const int r    = lane & 15;
    const int koff = (lane >> 4) << 1;           // 0 or 2

    const float* arow = E + (tm + r) * DIM + koff;
    const float* brow = E + (tn + r) * DIM + koff;

    v8f acc = {};
#pragma unroll
    for (int k = 0; k < DIM; k += 4) {
        v2f a = *(const v2f*)(arow + k);
        v2f b = *(const v2f*)(brow + k);
        acc = __builtin_amdgcn_wmma_f32_16x16x4_f32(
            /*neg_a=*/false, a, /*neg_b=*/false, b,
            /*c_mod=*/(short)0, acc, /*reuse_a=*/false, /*reuse_b=*/false);
    }

    const int n     = tn + (lane & 15);
    const int mbase = tm + ((lane >> 4) << 3);
    const float sqn = sq[n];
#pragma unroll
    for (int rr = 0; rr < 8; ++rr) {
        int m = mbase + rr;
        float d2 = sq[m] + sqn - 2.0f * acc[rr];
        dist[m * BATCH + n] = sqrtf(fmaxf(d2, 0.0f));
    }
}

// ---------------------------------------------------------------------------
// Kernel 3: hard-negative mining (argmin over negatives per row)
//   One wave32 per row; butterfly shuffle argmin (lowest-index tiebreak).
// ---------------------------------------------------------------------------
__global__ __launch_bounds__(256) void hard_kernel(const float* __restrict__ dist,
                                                   const int* __restrict__ labels,
                                                   int* __restrict__ hard_idx,
                                                   int* __restrict__ has_neg) {
    int wave = (blockIdx.x * blockDim.x + threadIdx.x) >> 5;
    int lane = threadIdx.x & 31;
    if (wave >= BATCH) return;
    const int i  = wave;
    const int li = labels[i];
    const float* drow = dist + i * BATCH;

    float best = INFINITY;
    int   bidx = BATCH;  // sentinel > any index
    for (int j = lane; j < BATCH; j += 32) {
        if (labels[j] != li) {
            float d = drow[j];
            if (d < best || (d == best && j < bidx)) { best = d; bidx = j; }
        }
    }
#pragma unroll
    for (int off = 16; off > 0; off >>= 1) {
        float ob = __shfl_xor(best, off, 32);
        int   oi = __shfl_xor(bidx, off, 32);
        if (ob < best || (ob == best && oi < bidx)) { best = ob; bidx = oi; }
    }
    if (lane == 0) {
        int hn = (bidx < BATCH) ? 1 : 0;
        hard_idx[i] = hn ? bidx : 0;
        has_neg[i]  = hn;
    }
}

// ---------------------------------------------------------------------------
// Kernel 4: per-p semi-hard selection + loss partials.
//   Gumbel-argmax over candidates == uniform categorical pick, emulated with
//   a deterministic integer hash keyed on (p,i,j).
// ---------------------------------------------------------------------------
__device__ __forceinline__ unsigned mix_hash(unsigned x) {
    x ^= x >> 16; x *= 0x7feb352dU;
    x ^= x >> 15; x *= 0x846ca68bU;
    x ^= x >> 16;
    return x;
}

__global__ __launch_bounds__(256) void loss_kernel(const float* __restrict__ dist,
                                                   const int* __restrict__ labels,
                                                   const int* __restrict__ hard_idx,
                                                   const int* __restrict__ has_neg,
                                                   float* __restrict__ loss_p,
                                                   int* __restrict__ cnt_p) {
    const int p    = blockIdx.x;
    const int lane = threadIdx.x & 31;
    const int wave = threadIdx.x >> 5;       // 8 waves per block
    const int lp   = labels[p];
    const unsigned pseed = mix_hash(0x9e3779b9u ^ (unsigned)p);

    __shared__ float s_loss[8];
    __shared__ int   s_cnt[8];

    float wl = 0.f;
    int   wc = 0;

    for (int i = wave; i < BATCH; i += 8) {
        if (i == p) continue;
        if (labels[i] != lp) continue;       // pos_mask requires same label
        if (!has_neg[i]) continue;           // valid requires a negative

        const float* drow  = dist + i * BATCH;
        const float  pos_d = drow[p];
        const unsigned iseed = mix_hash(pseed ^ (unsigned)(i * 0x85ebca6b));

        // semi-hard window scan with random (hash) argmax
        unsigned bestk = 0;
        int      bidx  = -1;
        for (int j = lane; j < BATCH; j += 32) {
            if (labels[j] == lp) continue;   // need a negative
            float d = drow[j];
            if (d > pos_d && d < pos_d + 1.0f) {
                unsigned k = mix_hash(iseed + (unsigned)j);
                if (bidx < 0 || k > bestk) { bestk = k; bidx = j; }
            }
        }
#pragma unroll
        for (int off = 16; off > 0; off >>= 1) {
            unsigned ok = __shfl_xor(bestk, off, 32);
            int      oi = __shfl_xor(bidx, off, 32);
            if (oi >= 0 && (bidx < 0 || ok > bestk)) { bestk = ok; bidx = oi; }
        }

        if (lane == 0) {
            int idx = (bidx >= 0) ? bidx : hard_idx[i];
            float neg_d = drow[idx];
            float tl = pos_d - neg_d + 1.0f;   // margin = 1.0
            if (tl > 0.f) { wl += tl; wc += 1; }
        }
    }

    if (lane == 0) { s_loss[wave] = wl; s_cnt[wave] = wc; }
    __syncthreads();
    if (threadIdx.x == 0) {
        float l = 0.f; int c = 0;
#pragma unroll
        for (int w = 0; w < 8; ++w) { l += s_loss[w]; c += s_cnt[w]; }
        loss_p[p] = l;
        cnt_p[p]  = c;
    }
}

// ---------------------------------------------------------------------------
// Kernel 5: deterministic final reduction
// ---------------------------------------------------------------------------
__global__ __launch_bounds__(256) void finalize_kernel(const float* __restrict__ loss_p,
                                                       const int* __restrict__ cnt_p,
                                                       float* __restrict__ out) {
    __shared__ float sl[256];
    __shared__ int   sc[256];
    int t = threadIdx.x;
    sl[t] = loss_p[t] + loss_p[t + 256];
    sc[t] = cnt_p[t] + cnt_p[t + 256];
    __syncthreads();
#pragma unroll
    for (int s = 128; s > 0; s >>= 1) {
        if (t < s) { sl[t] += sl[t + s]; sc[t] += sc[t + s]; }
        __syncthreads();
    }
    if (t == 0) out[0] = (sc[0] != 0) ? (sl[0] / (float)sc[0]) : 0.0f;
}

// ---------------------------------------------------------------------------
extern "C" void kernel_launch(void* const* d_in, const int* in_sizes, int n_in,
                              void* d_out, int out_size, void* d_ws, size_t ws_size,
                              hipStream_t stream) {
    const float* E      = (const float*)d_in[0];
    const int*   labels = (const int*)d_in[1];
    float*       out    = (float*)d_out;

    char* ws = (char*)d_ws;
    size_t off = 0;
    float* sq       = (float*)(ws + off); off += BATCH * sizeof(float);          // 2 KB
    float* dist     = (float*)(ws + off); off += (size_t)BATCH * BATCH * sizeof(float); // 1 MB
    int*   hard_idx = (int*)(ws + off);   off += BATCH * sizeof(int);
    int*   has_neg  = (int*)(ws + off);   off += BATCH * sizeof(int);
    float* loss_p   = (float*)(ws + off); off += BATCH * sizeof(float);
    int*   cnt_p    = (int*)(ws + off);   off += BATCH * sizeof(int);

    sq_kernel<<<2, 256, 0, stream>>>(E, sq);
    gram_dist_kernel<<<128, 256, 0, stream>>>(E, sq, dist);     // 1024 waves, 1 tile each
    hard_kernel<<<64, 256, 0, stream>>>(dist, labels, hard_idx, has_neg);
    loss_kernel<<<BATCH, 256, 0, stream>>>(dist, labels, hard_idx, has_neg, loss_p, cnt_p);
    finalize_kernel<<<1, 256, 0, stream>>>(loss_p, cnt_p, out);
}